// PointPillarsVoxelization_14843406975494
// MI455X (gfx1250) — compile-verified
//
#include <hip/hip_runtime.h>
#include <stdint.h>

// ---- problem constants (match reference) ----
#define NXc   440
#define NYc   500
#define NZc   1
#define Gc    (NXc * NYc * NZc)      // 220000 cells
#define Nc    2000000
#define CHc   5
#define MAXV  40000
#define MAXP  32
#define SCANB 1024
#define NSCAN ((Gc + SCANB - 1) / SCANB)   // 215 scan blocks

// out layout (floats): voxels [0, 6.4M) | coords [6.4M, 6.52M) | num [6.52M, 6.56M)
#define OUT_VOX_ELEMS ((size_t)MAXV * MAXP * CHc)   // 6,400,000
#define OUT_CRD_ELEMS ((size_t)MAXV * 3)            //   120,000

__device__ __forceinline__ int cell_of(float x, float y, float z, bool* valid) {
    const float rminx = 0.0f, rminy = -40.0f, rminz = -3.0f;
    const float rmaxx = 70.4f, rmaxy = 40.0f, rmaxz = 1.0f;
    bool v = (x >= rminx) && (x < rmaxx) &&
             (y >= rminy) && (y < rmaxy) &&
             (z >= rminz) && (z < rmaxz);
    *valid = v;
    int cx = (int)floorf((x - rminx) / 0.16f);
    int cy = (int)floorf((y - rminy) / 0.16f);
    int cz = (int)floorf((z - rminz) / 4.0f);
    cx = min(max(cx, 0), NXc - 1);
    cy = min(max(cy, 0), NYc - 1);
    cz = min(max(cz, 0), NZc - 1);
    return (cz * NYc + cy) * NXc + cx;
}

// 1) zero d_out + cell histogram + per-voxel slot counters (idempotent per call)
__global__ void k_init(float* __restrict__ out, int out_size,
                       unsigned* __restrict__ cnt, unsigned* __restrict__ vcnt) {
    int total = out_size + Gc + MAXV;
    for (int i = blockIdx.x * blockDim.x + threadIdx.x; i < total;
         i += gridDim.x * blockDim.x) {
        if (i < out_size)            out[i] = 0.0f;
        else if (i < out_size + Gc)  cnt[i - out_size] = 0u;
        else                         vcnt[i - out_size - Gc] = 0u;
    }
}

// 2) streaming pass: 4 points/thread -> five aligned float4 (b128) loads.
//    Histogram cells with L2-resident atomics; persist per-point cell id so the
//    scatter pass never re-reads the 40MB point stream.
__global__ void k_count(const float* __restrict__ pts, unsigned* __restrict__ cnt,
                        int* __restrict__ lins) {
    int q = blockIdx.x * blockDim.x + threadIdx.x;   // quad-of-points index
    int base = q * 4;
    if (base >= Nc) return;
    const float4* p4 = (const float4*)(pts + (size_t)base * CHc);
    __builtin_prefetch(p4 + 5 * 4096, 0, 0);
    float4 f0 = p4[0], f1 = p4[1], f2 = p4[2], f3 = p4[3], f4 = p4[4];
    // point j xyz = floats e[5j..5j+2] of the 20-float window
    float xs[4] = {f0.x, f1.y, f2.z, f3.w};
    float ys[4] = {f0.y, f1.z, f2.w, f4.x};
    float zs[4] = {f0.z, f1.w, f3.x, f4.y};
    int l[4];
#pragma unroll
    for (int j = 0; j < 4; ++j) {
        bool v; int lin = cell_of(xs[j], ys[j], zs[j], &v);
        l[j] = v ? lin : -1;
        if (v) atomicAdd(&cnt[lin], 1u);
    }
    *(int4*)(lins + base) = make_int4(l[0], l[1], l[2], l[3]);  // aligned b128 store
}

// 3a) per-block occupancy sums (wave32 ballot reduce)
__global__ void k_bsum(const unsigned* __restrict__ cnt, int* __restrict__ bsums) {
    __shared__ int wsums[32];
    int c = blockIdx.x * SCANB + threadIdx.x;
    int flag = (c < Gc && cnt[c] != 0u) ? 1 : 0;
    unsigned m = (unsigned)__ballot(flag);
    int lane = threadIdx.x & 31, wid = threadIdx.x >> 5;
    if (lane == 0) wsums[wid] = __popc(m);
    __syncthreads();
    if (wid == 0) {
        int s = wsums[lane];
        for (int d = 1; d < 32; d <<= 1) {
            int n = __shfl_up(s, d, 32);
            if (lane >= d) s += n;
        }
        if (lane == 31) bsums[blockIdx.x] = s;
    }
}

// 3b) single-block exclusive scan of the 215 block sums.
//     Pull them into LDS via the gfx1250 async-to-LDS path (ASYNCcnt).
__global__ void k_scantop(int* __restrict__ bsums) {
    __shared__ int sm[256];
    int t = threadIdx.x;
    if (t >= NSCAN) {
        sm[t] = 0;                     // no overlap with async-written slots
    } else {
        unsigned lds_off = (unsigned)(uintptr_t)(&sm[t]);   // LDS byte offset
        unsigned voff = (unsigned)t * 4u;                   // byte offset vs saddr
        asm volatile("global_load_async_to_lds_b32 %0, %1, %2"
                     :: "v"(lds_off), "v"(voff), "s"(bsums) : "memory");
    }
    asm volatile("s_wait_asynccnt 0x0" ::: "memory");
    __syncthreads();
    int v = sm[t];
    for (int d = 1; d < 256; d <<= 1) {
        int n = (t >= d) ? sm[t - d] : 0;
        __syncthreads();
        sm[t] += n;
        __syncthreads();
    }
    if (t < NSCAN) bsums[t] = sm[t] - v;   // exclusive
}

// 3c) assign voxel ids to cells; emit coords + num_points for first MAXV voxels
__global__ void k_apply(const unsigned* __restrict__ cnt, const int* __restrict__ bsums,
                        int* __restrict__ vid, float* __restrict__ out) {
    __shared__ int wsums[32];
    int c = blockIdx.x * SCANB + threadIdx.x;
    int lane = threadIdx.x & 31, wid = threadIdx.x >> 5;
    unsigned cc = (c < Gc) ? cnt[c] : 0u;
    int flag = (cc != 0u) ? 1 : 0;
    unsigned m = (unsigned)__ballot(flag);
    int wexcl = __popc(m & ((1u << lane) - 1u));
    if (lane == 0) wsums[wid] = __popc(m);
    __syncthreads();
    if (wid == 0) {
        int v0 = wsums[lane];
        int s = v0;
        for (int d = 1; d < 32; d <<= 1) {
            int n = __shfl_up(s, d, 32);
            if (lane >= d) s += n;
        }
        wsums[lane] = s - v0;   // exclusive per-wave offset
    }
    __syncthreads();
    if (c < Gc) {
        int out_id = -1;
        if (flag) {
            int v = bsums[blockIdx.x] + wsums[wid] + wexcl;
            if (v < MAXV) {
                out_id = v;
                int cx = c % NXc;
                int r  = c / NXc;
                int cy = r % NYc;
                int cz = r / NYc;
                float* coords = out + OUT_VOX_ELEMS;
                coords[v * 3 + 0] = (float)cz;
                coords[v * 3 + 1] = (float)cy;
                coords[v * 3 + 2] = (float)cx;
                float* nump = out + OUT_VOX_ELEMS + OUT_CRD_ELEMS;
                nump[v] = (float)min((int)cc, MAXP);
            }
        }
        vid[c] = out_id;
    }
}

// 4) per-point: claim a slot in its voxel, record original point index.
//    Reads only the 4B/point cell-id stream (no xyz re-read).
__global__ void k_scatter(const int* __restrict__ lins, const int* __restrict__ vid,
                          unsigned* __restrict__ vcnt, int* __restrict__ slots) {
    int i = blockIdx.x * blockDim.x + threadIdx.x;
    if (i >= Nc) return;
    __builtin_prefetch(lins + i + 16384, 0, 0);
    int lin = lins[i];
    if (lin < 0) return;
    int w = vid[lin];
    if (w < 0) return;
    unsigned slot = atomicAdd(&vcnt[w], 1u);
    if (slot < MAXP) slots[w * MAXP + slot] = i;
}

// 5) one wave32 per voxel: bitonic-sort 32 slot indices ascending (restores the
//    reference's stable original-index ordering), gather features, then stage
//    the 32x5 tile in LDS and write 160 contiguous floats coalesced.
__global__ void k_sortwrite(const float* __restrict__ pts,
                            const unsigned* __restrict__ vcnt,
                            const int* __restrict__ slots,
                            float* __restrict__ out) {
    __shared__ float stage[8][MAXP * CHc];   // 8 waves x 160 floats
    int wid = threadIdx.x >> 5, lane = threadIdx.x & 31;
    int v = blockIdx.x * (blockDim.x >> 5) + wid;
    bool active = (v < MAXV);
    int key = 0x7FFFFFFF;
    if (active) {
        int n = min((int)vcnt[v], MAXP);
        key = (lane < n) ? slots[v * MAXP + lane] : 0x7FFFFFFF;
    }
    // 32-lane bitonic sort, ascending
#pragma unroll
    for (int k = 2; k <= 32; k <<= 1) {
#pragma unroll
        for (int j = k >> 1; j >= 1; j >>= 1) {
            int p = __shfl_xor(key, j, 32);
            bool lower = (lane & j) == 0;
            bool asc   = (lane & k) == 0;
            int mn = min(key, p), mx = max(key, p);
            key = (lower == asc) ? mn : mx;
        }
    }
    float f[CHc] = {0.f, 0.f, 0.f, 0.f, 0.f};
    if (key != 0x7FFFFFFF) {
        const float* src = pts + (size_t)key * CHc;
#pragma unroll
        for (int c2 = 0; c2 < CHc; ++c2) f[c2] = src[c2];
    }
#pragma unroll
    for (int c2 = 0; c2 < CHc; ++c2) stage[wid][lane * CHc + c2] = f[c2];
    __syncthreads();
    if (active) {
        float* dst = out + (size_t)v * (MAXP * CHc);
#pragma unroll
        for (int k = 0; k < CHc; ++k)
            dst[lane + 32 * k] = stage[wid][lane + 32 * k];
    }
}

extern "C" void kernel_launch(void* const* d_in, const int* in_sizes, int n_in,
                              void* d_out, int out_size, void* d_ws, size_t ws_size,
                              hipStream_t stream) {
    (void)in_sizes; (void)n_in; (void)ws_size;
    const float* pts = (const float*)d_in[0];
    float* out = (float*)d_out;

    // workspace carve-up (4-byte words): ~16 MB total
    unsigned* cnt   = (unsigned*)d_ws;                 // [Gc]
    int*      vid   = (int*)(cnt + Gc);                // [Gc]
    unsigned* vcnt  = (unsigned*)(vid + Gc);           // [MAXV]
    int*      slots = (int*)(vcnt + MAXV);             // [MAXV*MAXP]
    int*      bsums = slots + (size_t)MAXV * MAXP;     // [256]
    int*      lins  = bsums + 256;                     // [Nc]

    const int TB = 256;
    const int quadBlocks = (Nc / 4 + TB - 1) / TB;     // 1954
    const int ptBlocks   = (Nc + TB - 1) / TB;         // 7813

    k_init<<<2048, TB, 0, stream>>>(out, out_size, cnt, vcnt);
    k_count<<<quadBlocks, TB, 0, stream>>>(pts, cnt, lins);
    k_bsum<<<NSCAN, SCANB, 0, stream>>>(cnt, bsums);
    k_scantop<<<1, 256, 0, stream>>>(bsums);
    k_apply<<<NSCAN, SCANB, 0, stream>>>(cnt, bsums, vid, out);
    k_scatter<<<ptBlocks, TB, 0, stream>>>(lins, vid, vcnt, slots);
    k_sortwrite<<<(MAXV * 32 + TB - 1) / TB, TB, 0, stream>>>(pts, vcnt, slots, out);
}